// StructNConv2D_d_with_s_39281770889957
// MI455X (gfx1250) — compile-verified
//
#include <hip/hip_runtime.h>

typedef __attribute__((ext_vector_type(2))) float v2f;
typedef __attribute__((ext_vector_type(8))) float v8f;

#define EPSF 1e-20f
#define CH   16
#define HH   256
#define WW   256
#define HW   (HH * WW)
#define NPIX (4 * HW)          // B * H * W = 262144 flattened pixels

__device__ __forceinline__ float softplus_f(float x) {
    // numerically stable softplus: max(x,0) + log1p(exp(-|x|))
    return fmaxf(x, 0.0f) + log1pf(expf(-fabsf(x)));
}

// d_ws float layout: [0..143] sw (C,9) ; [144..399] cw (O,I) ; [400] sum_sw ; [401] sum_cw
__global__ void sncv_setup_kernel(const float* __restrict__ spatial_weight,
                                  const float* __restrict__ channel_weight,
                                  float* __restrict__ ws) {
    int t = threadIdx.x;
    if (t < CH * 9)   ws[t]       = softplus_f(spatial_weight[t]);   // (C,1,9,1,1) flat = c*9+k
    if (t < CH * CH)  ws[144 + t] = softplus_f(channel_weight[t]);   // (O,I,1,1)  flat = o*16+i
    __syncthreads();
    if (t == 0) { float s = 0.f; for (int i = 0; i < CH * 9;  ++i) s += ws[i];        ws[400] = s; }
    if (t == 1) { float s = 0.f; for (int i = 0; i < CH * CH; ++i) s += ws[144 + i];  ws[401] = s; }
}

// Spatial normalized 3x3 circular conv for one (channel, pixel).
// Produces the two B-operand values for the channel-stage WMMAs:
//   bnom = cd_spatial * d_spatial,  bden = cd_spatial
__device__ __forceinline__ void sncv_spatial(const float* __restrict__ dP,
                                             const float* __restrict__ cdP,
                                             const float* __restrict__ sprP,
                                             const float* __restrict__ sw,
                                             float inv_sum_sw,
                                             int b, int c, int h, int w,
                                             float& bnom, float& bden) {
    const float* dch  = dP  + ((size_t)(b * CH + c)) * HW;
    const float* cdch = cdP + ((size_t)(b * CH + c)) * HW;
    const float* sprc = sprP + ((size_t)(b * CH + c)) * 9 * HW + h * WW + w;
    const float* swc  = sw + c * 9;
    float nom = 0.0f, den = 0.0f;
#pragma unroll
    for (int i = 0; i < 3; ++i) {
        // jnp.roll(x, 1-i, axis=-2): rolled[h] = x[(h - (1-i)) mod H]
        const int hh = (h - (1 - i)) & (HH - 1);
#pragma unroll
        for (int j = 0; j < 3; ++j) {
            const int ww = (w - (1 - j)) & (WW - 1);
            const int k  = i * 3 + j;
            const float dv = dch[hh * WW + ww];
            const float cv = cdch[hh * WW + ww];
            const float sp = sprc[k * HW];
            const float cp = cv * sp * swc[k];   // cd_prop * sw
            nom = fmaf(cp, dv, nom);
            den += cp;
        }
    }
    const float d_sp  = nom / (den + EPSF);      // d_spatial
    const float cd_sp = den * inv_sum_sw;        // cd_spatial
    bnom = cd_sp * d_sp;
    bden = cd_sp;
}

// One wave = 16 pixels x all 16 channels. Channel mixing done with chained
// V_WMMA_F32_16X16X4_F32 (4 chunks of K=4 -> K=16), two accumulators
// (nom2, den2). Per documented f32 A(16x4)/B(4x16) layouts:
//   lanes 0-15 carry K = {0,1} of each chunk, lanes 16-31 carry K = {2,3};
//   pixel/out-row index = lane % 16; D VGPR v -> rows v (lanes 0-15) / v+8.
__global__ __launch_bounds__(256) void sncv_fused_kernel(const float* __restrict__ dP,
                                                         const float* __restrict__ cdP,
                                                         const float* __restrict__ sprP,
                                                         const float* __restrict__ biasP,
                                                         const float* __restrict__ ws,
                                                         float* __restrict__ d_outP,
                                                         float* __restrict__ cd_outP) {
    const float* sw = ws;
    const float* cw = ws + 144;
    const float inv_sum_sw = 1.0f / ws[400];
    const float inv_sum_cw = 1.0f / ws[401];

    const int lane = threadIdx.x & 31;
    const int wave = threadIdx.x >> 5;
    const int p    = blockIdx.x * 128 + wave * 16 + (lane & 15);  // flattened (b,h,w)
    const int b    = p >> 16;
    const int h    = (p >> 8) & (HH - 1);
    const int w    = p & (WW - 1);

    const int kbase = (lane < 16) ? 0 : 2;   // which K pair this lane supplies
    const int m     = lane & 15;             // A-matrix row = output channel

    v8f acc_nom = {};
    v8f acc_den = {};
#pragma unroll
    for (int t = 0; t < 4; ++t) {
        const int c0 = 4 * t + kbase;

        float bn0, bd0, bn1, bd1;
        sncv_spatial(dP, cdP, sprP, sw, inv_sum_sw, b, c0,     h, w, bn0, bd0);
        sncv_spatial(dP, cdP, sprP, sw, inv_sum_sw, b, c0 + 1, h, w, bn1, bd1);

        v2f a;  a.x = cw[m * CH + c0];  a.y = cw[m * CH + c0 + 1];   // cw[o, i]
        v2f bn; bn.x = bn0; bn.y = bn1;
        v2f bd; bd.x = bd0; bd.y = bd1;

        acc_nom = __builtin_amdgcn_wmma_f32_16x16x4_f32(false, a, false, bn,
                                                        (short)0, acc_nom, false, false);
        acc_den = __builtin_amdgcn_wmma_f32_16x16x4_f32(false, a, false, bd,
                                                        (short)0, acc_den, false, false);
    }

    // D layout: VGPR v -> out channel o = v + (lane<16 ? 0 : 8), pixel = lane%16
    const int obase = (lane < 16) ? 0 : 8;
#pragma unroll
    for (int v = 0; v < 8; ++v) {
        const int o = obase + v;
        const float nom2 = acc_nom[v];
        const float den2 = acc_den[v];
        const float dv = nom2 / (den2 + EPSF) + biasP[o];
        const float cv = den2 * inv_sum_cw;
        const size_t idx = ((size_t)(b * CH + o)) * HW + h * WW + w;
        d_outP[idx]  = dv;
        cd_outP[idx] = cv;
    }
}

extern "C" void kernel_launch(void* const* d_in, const int* in_sizes, int n_in,
                              void* d_out, int out_size, void* d_ws, size_t ws_size,
                              hipStream_t stream) {
    (void)in_sizes; (void)n_in; (void)out_size; (void)ws_size;
    const float* dP   = (const float*)d_in[0];   // d   (4,16,256,256)
    const float* cdP  = (const float*)d_in[1];   // cd
    // d_in[2] = s, d_in[3] = cs : unused by the reference computation
    const float* sprP = (const float*)d_in[4];   // s_prod_roll (4,16,9,256,256)
    const float* swP  = (const float*)d_in[5];   // spatial_weight (16,1,9,1,1)
    const float* cwP  = (const float*)d_in[6];   // channel_weight (16,16,1,1)
    const float* bsP  = (const float*)d_in[7];   // bias (1,16,1,1)

    float* outP   = (float*)d_out;
    float* d_outP  = outP;                         // first output:  d_out
    float* cd_outP = outP + (size_t)4 * CH * HW;   // second output: cd_out
    float* ws = (float*)d_ws;

    sncv_setup_kernel<<<1, 256, 0, stream>>>(swP, cwP, ws);

    const int blocks = NPIX / 128;   // 128 pixels per 256-thread block (8 waves x 16 px)
    sncv_fused_kernel<<<blocks, 256, 0, stream>>>(dP, cdP, sprP, bsP, ws, d_outP, cd_outP);
}